// Linear_48825188221071
// MI455X (gfx1250) — compile-verified
//
#include <hip/hip_runtime.h>
#include <math.h>

// irreps: 128x0e + 64x1o + 32x2e + 16x3o ; DIM = 592, N = 100000
// Per irrep l: y[n,w,i] = (1/sqrt(mul)) * sum_u W[u,w] * x[n, off + u*ird + i]
// V_WMMA_F32_16X16X4_F32 with 2x2 register blocking: each wave computes a
// 32n x (16*NW)w output macro-tile, reusing A fragments across w-tiles and
// B fragments across n-tiles (halves VMEM instructions per WMMA).

typedef __attribute__((ext_vector_type(2))) float v2f;
typedef __attribute__((ext_vector_type(8))) float v8f;

static constexpr int kDIM = 592;
static constexpr int kN = 100000;
static constexpr int kNTILES = kN / 16;          // 6250, exact
static constexpr int kNT2 = kNTILES / 2;         // 3125, exact (32-row macro-tiles)
static constexpr int kWavesPerBlock = 8;         // 256 threads, wave32

template <int MUL, int IRD, int OFF, int NW>
__global__ __launch_bounds__(256) void eqlin_wmma_kernel(
    const float* __restrict__ feat,   // [N, 592]
    const float* __restrict__ w,      // [MUL, MUL] row-major
    float* __restrict__ out,          // [N, 592]
    float alpha) {
  const int lane = threadIdx.x & 31;
  const int wave = threadIdx.x >> 5;
  const int nt2 = blockIdx.x * kWavesPerBlock + wave;
  if (nt2 >= kNT2) return;                       // wave-uniform exit: EXEC all-1s for WMMA

  const int wtile = blockIdx.y / IRD;            // (16*NW)-wide output-column tile
  const int comp  = blockIdx.y % IRD;            // which of the 2l+1 components

  const int half = lane >> 4;                    // 0: lanes 0-15, 1: lanes 16-31
  const int l16  = lane & 15;
  const int n0    = nt2 * 32;
  const int wbase = wtile * (16 * NW);

  // A fragments: lane holds X[row, u=k+2*half] (VGPR0) and u=k+2*half+1 (VGPR1)
  const float* aptr0 = feat + (size_t)(n0 + l16) * kDIM + OFF + comp + (2 * half) * IRD;
  const float* aptr1 = aptr0 + (size_t)16 * kDIM;
  // B fragments: lane holds W[k+2*half, wbase+j*16+l16] / W[k+2*half+1, ...]
  const float* bptr = w + (2 * half) * MUL + wbase + l16;

  v8f c[2][NW];
#pragma unroll
  for (int i = 0; i < 2; ++i)
#pragma unroll
    for (int j = 0; j < NW; ++j) c[i][j] = v8f{};

#pragma unroll 8
  for (int k = 0; k < MUL; k += 4) {
    v2f a0, a1, b[NW];
    if constexpr (IRD == 1) {
      // consecutive u contiguous -> single 8B load per fragment (8B aligned)
      a0 = *reinterpret_cast<const v2f*>(aptr0 + k);
      a1 = *reinterpret_cast<const v2f*>(aptr1 + k);
    } else {
      a0.x = aptr0[(k + 0) * IRD];
      a0.y = aptr0[(k + 1) * IRD];
      a1.x = aptr1[(k + 0) * IRD];
      a1.y = aptr1[(k + 1) * IRD];
    }
#pragma unroll
    for (int j = 0; j < NW; ++j) {
      b[j].x = bptr[(k + 0) * MUL + j * 16];
      b[j].y = bptr[(k + 1) * MUL + j * 16];
    }
#pragma unroll
    for (int j = 0; j < NW; ++j) {
      c[0][j] = __builtin_amdgcn_wmma_f32_16x16x4_f32(
          false, a0, false, b[j], (short)0, c[0][j], false, false);
      c[1][j] = __builtin_amdgcn_wmma_f32_16x16x4_f32(
          false, a1, false, b[j], (short)0, c[1][j], false, false);
    }
  }

  // D layout: VGPR r -> row (tile_base + r + 8*half), col wbase + j*16 + l16
#pragma unroll
  for (int i = 0; i < 2; ++i) {
    float* optr = out + (size_t)(n0 + i * 16 + 8 * half) * kDIM + OFF +
                  (wbase + l16) * IRD + comp;
#pragma unroll
    for (int j = 0; j < NW; ++j) {
#pragma unroll
      for (int r = 0; r < 8; ++r) {
        optr[(size_t)r * kDIM + (size_t)(j * 16) * IRD] = c[i][j][r] * alpha;
      }
    }
  }
}

extern "C" void kernel_launch(void* const* d_in, const int* in_sizes, int n_in,
                              void* d_out, int out_size, void* d_ws, size_t ws_size,
                              hipStream_t stream) {
  (void)in_sizes; (void)n_in; (void)out_size; (void)d_ws; (void)ws_size;
  const float* feat = (const float*)d_in[0];
  const float* w0   = (const float*)d_in[1];   // 128x128
  const float* w1   = (const float*)d_in[2];   // 64x64
  const float* w2   = (const float*)d_in[3];   // 32x32
  const float* w3   = (const float*)d_in[4];   // 16x16
  float* out = (float*)d_out;

  const int gx = (kNT2 + kWavesPerBlock - 1) / kWavesPerBlock;  // 391
  dim3 block(256);

  // offsets: l0 @0 (128*1), l1 @128 (64*3), l2 @320 (32*5), l3 @480 (16*7); 480+112=592
  eqlin_wmma_kernel<128, 1,   0, 2><<<dim3(gx, (128 / 32) * 1), block, 0, stream>>>(
      feat, w0, out, 1.0f / sqrtf(128.0f));
  eqlin_wmma_kernel< 64, 3, 128, 2><<<dim3(gx, ( 64 / 32) * 3), block, 0, stream>>>(
      feat, w1, out, 1.0f / sqrtf(64.0f));
  eqlin_wmma_kernel< 32, 5, 320, 2><<<dim3(gx, ( 32 / 32) * 5), block, 0, stream>>>(
      feat, w2, out, 1.0f / sqrtf(32.0f));
  eqlin_wmma_kernel< 16, 7, 480, 1><<<dim3(gx, ( 16 / 16) * 7), block, 0, stream>>>(
      feat, w3, out, 1.0f / sqrtf(16.0f));
}